// HeightRecovery_76373108457573
// MI455X (gfx1250) — compile-verified
//
#include <hip/hip_runtime.h>
#include <hip/hip_bf16.h>
#include <math.h>

namespace {
constexpr int H = 966;
constexpr int W = 1280;
constexpr int BX = 16;
constexpr int BY = 16;
constexpr int TW  = BX + 2;           // 18: halo tile width
constexpr int THT = BY + 2;           // 18: halo tile height
constexpr int TILE_ELEMS = TW * THT;  // 324
constexpr int PSTRIDE = TW + 1;       // 19 floats: odd stride, dodge LDS bank conflicts
constexpr int TILES_X = (W - 2 + BX - 1) / BX;   // 80
constexpr int TILES_Y = (H - 2 + BY - 1) / BY;   // 61
constexpr int NUM_TILES = TILES_X * TILES_Y;     // 4880
constexpr int TPB = 4;                            // tiles per persistent block
constexpr int NBLOCKS = (NUM_TILES + TPB - 1) / TPB; // 1220
}

struct F3 { float x, y, z; };

__device__ __forceinline__ F3 cross_normalize(const F3 a, const F3 b) {
    F3 r;
    r.x = a.y * b.z - a.z * b.y;
    r.y = a.z * b.x - a.x * b.z;
    r.z = a.x * b.y - a.y * b.x;
    float len = sqrtf(r.x * r.x + r.y * r.y + r.z * r.z);
    float s = 1.0f / fmaxf(len, 1e-12f);
    r.x *= s; r.y *= s; r.z *= s;
    return r;
}

__device__ __forceinline__ void tile_origin(int t, int& r0, int& c0, int& tr, int& tc) {
    const int ty = t / TILES_X, tx = t % TILES_X;
    r0 = 1 + ty * BY;
    c0 = 1 + tx * BX;
    tr = (r0 - 1 < H - THT) ? (r0 - 1) : (H - THT);   // clamp: 18x18 window in-bounds
    tc = (c0 - 1 < W - TW)  ? (c0 - 1) : (W - TW);
}

// ---------------- CDNA5 Tensor Data Mover path ------------------------------
#if defined(__HIP_DEVICE_COMPILE__) && __has_builtin(__builtin_amdgcn_tensor_load_to_lds)
#define USE_TDM 1
#else
#define USE_TDM 0
#endif

#if USE_TDM
typedef unsigned int u32x4 __attribute__((ext_vector_type(4)));
typedef int          i32x4 __attribute__((ext_vector_type(4)));
typedef int          i32x8 __attribute__((ext_vector_type(8)));

// DMA one 18x18 f32 tile (row stride = W elements) from global memory into LDS.
// D# layout per CDNA5 ISA 8.3/8.4. 2D tile: groups 2/3 zero, tile_dim2=0.
__device__ __forceinline__ void tdm_load_tile18(const float* gsrc, const float* ldsdst) {
    const uint64_t ga  = (uint64_t)(uintptr_t)gsrc;
    // Low 32 bits of a generic LDS address are the LDS byte offset (ISA 10.2).
    const uint32_t lds = (uint32_t)(uintptr_t)(const void*)ldsdst;

    u32x4 g0;
    g0[0] = 1u;                                                // count=1, user descriptor
    g0[1] = lds;                                               // lds_addr (bytes)
    g0[2] = (uint32_t)ga;                                      // global_addr[31:0]
    g0[3] = (uint32_t)((ga >> 32) & 0x01FFFFFFu) | (2u << 30); // addr[56:32] | type=2

    i32x8 g1;
    g1[0] = (int)(2u << 16);                                   // data_size=4B; wg_mask=0
    g1[1] = (int)(((uint32_t)W & 0xFFFFu) << 16);              // tensor_dim0[15:0]
    g1[2] = (int)(((uint32_t)W >> 16) | (((uint32_t)H & 0xFFFFu) << 16));
    g1[3] = (int)(((uint32_t)H >> 16) | ((uint32_t)TW << 16)); // dim1 hi | tile_dim0=18
    g1[4] = (int)THT;                                          // tile_dim1=18; tile_dim2=0
    g1[5] = (int)W;                                            // tensor_dim0_stride lo
    g1[6] = 0;
    g1[7] = 0;

    i32x4 z4 = {0, 0, 0, 0};
#if __clang_major__ >= 23
    i32x8 z8 = {0, 0, 0, 0, 0, 0, 0, 0};
    __builtin_amdgcn_tensor_load_to_lds(g0, g1, z4, z4, z8, 0);
#else
    __builtin_amdgcn_tensor_load_to_lds(g0, g1, z4, z4, 0);
#endif
}
#endif // USE_TDM

// ---------------- main fused kernel ----------------------------------------
// Persistent blocks: TPB consecutive tiles each, double-buffered TDM prefetch:
// while the stencil chews on tile t (LDS buf A), the Tensor Data Mover streams
// tile t+1 into buf B; wave 0 gates with s_wait_tensorcnt<=3 (TDM retires in
// order per wave, so <=3 outstanding == previous tile's 3 descriptors landed).
__global__ __launch_bounds__(BX * BY)
void hr_main(const float* __restrict__ norm, const float* __restrict__ theta,
             const float* __restrict__ aoi,  const float* __restrict__ mask,
             float* __restrict__ hout, float* __restrict__ nout) {
    __shared__ float s_raw[2][3][TILE_ELEMS];  // [buf][plane: n,t,a][elem]
    __shared__ float s_px[THT][PSTRIDE];
    __shared__ float s_py[THT][PSTRIDE];
    __shared__ float s_pz[THT][PSTRIDE];

    const int tid  = threadIdx.y * BX + threadIdx.x;
    const int t0   = blockIdx.x * TPB;
    const int tEnd = (t0 + TPB < NUM_TILES) ? (t0 + TPB) : NUM_TILES;
    if (t0 >= NUM_TILES) return;

#if USE_TDM
    if (threadIdx.y < 2) {                     // exactly wave 0 (wave32, 16x16 block)
        int r0, c0, tr, tc;
        tile_origin(t0, r0, c0, tr, tc);
        const size_t off = (size_t)tr * W + tc;
        tdm_load_tile18(norm  + off, s_raw[0][0]);
        tdm_load_tile18(theta + off, s_raw[0][1]);
        tdm_load_tile18(aoi   + off, s_raw[0][2]);
    }
#endif

    int cur = 0;
    for (int t = t0; t < tEnd; ++t) {
        int r0, c0, tr, tc;
        tile_origin(t, r0, c0, tr, tc);

#if USE_TDM
        if (threadIdx.y < 2) {
            if (t + 1 < tEnd) {
                int r0n, c0n, trn, tcn;
                tile_origin(t + 1, r0n, c0n, trn, tcn);
                const size_t offn = (size_t)trn * W + tcn;
                tdm_load_tile18(norm  + offn, s_raw[cur ^ 1][0]);
                tdm_load_tile18(theta + offn, s_raw[cur ^ 1][1]);
                tdm_load_tile18(aoi   + offn, s_raw[cur ^ 1][2]);
                __builtin_amdgcn_s_wait_tensorcnt(3);  // tile t's 3 descriptors done
            } else {
                __builtin_amdgcn_s_wait_tensorcnt(0);  // drain on last tile
            }
        }
#else
        for (int idx = tid; idx < TILE_ELEMS; idx += BX * BY) {
            const int g = (tr + idx / TW) * W + (tc + idx % TW);
            s_raw[cur][0][idx] = norm[g];
            s_raw[cur][1][idx] = theta[g];
            s_raw[cur][2][idx] = aoi[g];
        }
#endif
        __syncthreads();   // S1: raw[cur] valid for all waves; p-LDS free to overwrite

        // Phase 2: camera-space points p into LDS (recompute beats a 30 MB HBM
        // round-trip of the p array; 2 sincos per halo pixel is noise).
        for (int idx = tid; idx < TILE_ELEMS; idx += BX * BY) {
            const float nv = s_raw[cur][0][idx];
            float sa, ca, st, ct;
            sincosf(s_raw[cur][2][idx], &sa, &ca);
            sincosf(s_raw[cur][1][idx], &st, &ct);
            const float rw = sa * nv;
            const int rr = idx / TW, cc = idx % TW;
            s_px[rr][cc] = rw * ct;
            s_py[rr][cc] = rw * st;
            s_pz[rr][cc] = ca * nv;
        }
        __syncthreads();   // S2: p tile complete; raw[cur] reads all retired

        // Phase 3: 3x3 stencil -> averaged normal, height.
        const int r = r0 + threadIdx.y;
        const int c = c0 + threadIdx.x;
        if (r <= H - 2 && c <= W - 2) {
            const int lr = r - tr, lc = c - tc;
            const F3 ctr = {s_px[lr][lc], s_py[lr][lc], s_pz[lr][lc]};
            auto dv = [&](int dr, int dc) {
                F3 v;
                v.x = s_px[lr + dr][lc + dc] - ctr.x;
                v.y = s_py[lr + dr][lc + dc] - ctr.y;
                v.z = s_pz[lr + dr][lc + dc] - ctr.z;
                return v;
            };
            const F3 vx0 = dv(0, -1), vy0 = dv(-1, 0);
            const F3 vx1 = dv(0,  1), vy1 = dv( 1, 0);
            const F3 vaa = dv(-1, -1), vbb = dv(1, -1);   // v_x0y0, v_x0y1
            const F3 vcc = dv(-1,  1), vdd = dv(1,  1);   // v_x1y0, v_x1y1

            const F3 n0 = cross_normalize(vx0, vy0);
            const F3 n1 = cross_normalize(vx1, vy1);
            const F3 n2 = cross_normalize(vaa, vbb);
            const F3 n3 = cross_normalize(vcc, vdd);

            F3 s;
            s.x = (n0.x + n1.x + n2.x + n3.x) * 0.25f;
            s.y = (n0.y + n1.y + n2.y + n3.y) * 0.25f;
            s.z = (n0.z + n1.z + n2.z + n3.z) * 0.25f;
            const float len = sqrtf(s.x * s.x + s.y * s.y + s.z * s.z);
            const float sc = 1.0f / fmaxf(len, 1e-12f);
            s.x *= sc; s.y *= sc; s.z *= sc;

            const int g = r * W + c;
            hout[g] = fabsf(ctr.x * s.x + ctr.y * s.y + ctr.z * s.z) * mask[g];
            nout[3 * g + 0] = s.x;
            nout[3 * g + 1] = s.y;
            nout[3 * g + 2] = s.z;
        }
        cur ^= 1;
    }
}

// ---------------- border kernel: reflect-pad normals + border heights -------
__global__ void hr_border(const float* __restrict__ norm, const float* __restrict__ theta,
                          const float* __restrict__ aoi,  const float* __restrict__ mask,
                          float* __restrict__ hout, float* __restrict__ nout) {
    const int NB = 2 * W + 2 * (H - 2);
    const int idx = blockIdx.x * blockDim.x + threadIdx.x;
    if (idx >= NB) return;

    int r, c;
    if (idx < W)            { r = 0;      c = idx; }
    else if (idx < 2 * W)   { r = H - 1;  c = idx - W; }
    else { const int rem = idx - 2 * W; r = 1 + (rem >> 1); c = (rem & 1) ? (W - 1) : 0; }

    // jnp.pad(..., mode='reflect'): row 0 <- interior full-row 2, row H-1 <- H-3.
    const int ri = (r == 0) ? 2 : ((r == H - 1) ? H - 3 : r);
    const int ci = (c == 0) ? 2 : ((c == W - 1) ? W - 3 : c);

    const int gs = ri * W + ci;
    const float nx = nout[3 * gs + 0];
    const float ny = nout[3 * gs + 1];
    const float nz = nout[3 * gs + 2];

    const int g = r * W + c;
    const float nv = norm[g];
    float sa, ca, st, ct;
    sincosf(aoi[g], &sa, &ca);
    sincosf(theta[g], &st, &ct);
    const float rw = sa * nv;
    const float px = rw * ct, py = rw * st, pz = ca * nv;

    hout[g] = fabsf(px * nx + py * ny + pz * nz) * mask[g];
    nout[3 * g + 0] = nx;
    nout[3 * g + 1] = ny;
    nout[3 * g + 2] = nz;
}

extern "C" void kernel_launch(void* const* d_in, const int* in_sizes, int n_in,
                              void* d_out, int out_size, void* d_ws, size_t ws_size,
                              hipStream_t stream) {
    const float* norm  = (const float*)d_in[0];
    const float* theta = (const float*)d_in[1];
    const float* aoi   = (const float*)d_in[2];
    const float* mask  = (const float*)d_in[3];
    float* hout = (float*)d_out;                 // H*W heights
    float* nout = hout + (size_t)H * W;          // H*W*3 normals (HWC)

    dim3 block(BX, BY);
    hr_main<<<dim3(NBLOCKS), block, 0, stream>>>(norm, theta, aoi, mask, hout, nout);

    const int NB = 2 * W + 2 * (H - 2);
    hr_border<<<(NB + 255) / 256, 256, 0, stream>>>(norm, theta, aoi, mask, hout, nout);
}